// r_primal_old_62002147885374
// MI455X (gfx1250) — compile-verified
//
#include <hip/hip_runtime.h>
#include <stdint.h>

#define TPB  256            // 8 waves (wave32)
#define TILE 1024           // TPB * 4 elements per tile

typedef float vf4 __attribute__((ext_vector_type(4)));
typedef int   vi4 __attribute__((ext_vector_type(4)));

// ---- CDNA5 async global->LDS copy helpers (gfx1250) ----------------------
__device__ __forceinline__ uint32_t lds_off(const void* p) {
  // generic LDS pointer = {aperture_hi32, lds_offset32}; low 32 bits = LDS addr
  return (uint32_t)(uintptr_t)p;
}

__device__ __forceinline__ void async_b128(uint32_t lds, const void* gaddr) {
  // 16B per lane, non-temporal so the 240MB stream doesn't thrash L2 (x stays resident)
  asm volatile("global_load_async_to_lds_b128 %0, %1, off th:TH_LOAD_NT"
               :: "v"(lds), "v"(gaddr) : "memory");
}

__device__ __forceinline__ void wait_async_le3() {
  asm volatile("s_wait_asynccnt 3" ::: "memory");
}
__device__ __forceinline__ void wait_async_0() {
  asm volatile("s_wait_asynccnt 0" ::: "memory");
}

// hardware f32 atomic add, no return (STOREcnt), device scope
__device__ __forceinline__ void atomic_add_f32_dev(float* p, float v) {
  asm volatile("global_atomic_add_f32 %0, %1, off scope:SCOPE_DEV"
               :: "v"(p), "v"(v) : "memory");
}

// ---- kernel 1: COO SpMV scatter, async-DMA double buffered ---------------
// (placed first in the file so the disasm snippet shows the async path)
__global__ void __launch_bounds__(TPB)
spmv_scatter_kernel(const float* __restrict__ A_vals,
                    const int*   __restrict__ A_rows,
                    const int*   __restrict__ A_cols,
                    const float* __restrict__ x,
                    float*       __restrict__ Ax,
                    int nnz, int fullTiles) {
  __shared__ float s_vals[2][TILE];
  __shared__ int   s_rows[2][TILE];
  __shared__ int   s_cols[2][TILE];

  const int tid  = threadIdx.x;
  const int slot = tid * 4;   // each thread owns a private 16B slot per array

  const uint32_t lv[2] = { lds_off(&s_vals[0][slot]), lds_off(&s_vals[1][slot]) };
  const uint32_t lr[2] = { lds_off(&s_rows[0][slot]), lds_off(&s_rows[1][slot]) };
  const uint32_t lc[2] = { lds_off(&s_cols[0][slot]), lds_off(&s_cols[1][slot]) };

  int t = blockIdx.x;
  const int stride = gridDim.x;

  // prologue: DMA first tile into buffer 0
  if (t < fullTiles) {
    size_t base = (size_t)t * TILE + slot;
    async_b128(lv[0], A_vals + base);
    async_b128(lr[0], A_rows + base);
    async_b128(lc[0], A_cols + base);
  }

  int buf = 0;
  for (; t < fullTiles; t += stride) {
    int tn = t + stride;
    if (tn < fullTiles) {
      // prefetch next tile into the other buffer, then wait only for the
      // previous 3 DMAs (async loads complete in order)
      size_t base = (size_t)tn * TILE + slot;
      async_b128(lv[buf ^ 1], A_vals + base);
      async_b128(lr[buf ^ 1], A_rows + base);
      async_b128(lc[buf ^ 1], A_cols + base);
      wait_async_le3();
    } else {
      wait_async_0();
    }

    // consume own 16B slots as single ds_load_b128s:
    // no cross-thread LDS sharing -> no barrier needed
    vf4 v = *(const vf4*)&s_vals[buf][slot];
    vi4 c = *(const vi4*)&s_cols[buf][slot];
    vi4 r = *(const vi4*)&s_rows[buf][slot];
#pragma unroll
    for (int j = 0; j < 4; ++j) {
      atomic_add_f32_dev(Ax + r[j], v[j] * x[c[j]]);  // x gather: L2-resident
    }
    buf ^= 1;
  }

  // tail (nnz % TILE elements), handled by block 0 with plain loads
  if (blockIdx.x == 0) {
    for (int i = fullTiles * TILE + tid; i < nnz; i += TPB) {
      atomic_add_f32_dev(Ax + A_rows[i], A_vals[i] * x[A_cols[i]]);
    }
  }
}

// ---- kernel 2: proj = y + Iy*relu(-y); block-reduced inf-norms -----------
__global__ void __launch_bounds__(TPB)
proj_max_kernel(const float* __restrict__ Ax,
                const float* __restrict__ b,
                const float* __restrict__ Iy,
                unsigned* __restrict__ cells, int m) {
  float mp = 0.0f, mb = 0.0f;
  const int idx = blockIdx.x * TPB + threadIdx.x;
  const int stride = gridDim.x * TPB;

  const int m4 = m >> 2;
  const vf4* Ax4 = (const vf4*)Ax;
  const vf4* b4  = (const vf4*)b;
  const vf4* Iy4 = (const vf4*)Iy;
  for (int i = idx; i < m4; i += stride) {
    vf4 a  = __builtin_nontemporal_load(Ax4 + i);
    vf4 bb = __builtin_nontemporal_load(b4 + i);
    vf4 iy = __builtin_nontemporal_load(Iy4 + i);
#pragma unroll
    for (int j = 0; j < 4; ++j) {
      float y    = a[j] - bb[j];
      float proj = y + iy[j] * fmaxf(-y, 0.0f);
      mp = fmaxf(mp, fabsf(proj));
      mb = fmaxf(mb, fabsf(bb[j]));
    }
  }
  for (int i = m4 * 4 + idx; i < m; i += stride) {  // tail
    float y    = Ax[i] - b[i];
    float proj = y + Iy[i] * fmaxf(-y, 0.0f);
    mp = fmaxf(mp, fabsf(proj));
    mb = fmaxf(mb, fabsf(b[i]));
  }

  // wave32 reduction
#pragma unroll
  for (int off = 16; off > 0; off >>= 1) {
    mp = fmaxf(mp, __shfl_xor(mp, off, 32));
    mb = fmaxf(mb, __shfl_xor(mb, off, 32));
  }
  __shared__ float red[2][TPB / 32];
  const int wave = threadIdx.x >> 5, lane = threadIdx.x & 31;
  if (lane == 0) { red[0][wave] = mp; red[1][wave] = mb; }
  __syncthreads();
  if (threadIdx.x == 0) {
#pragma unroll
    for (int w = 1; w < TPB / 32; ++w) {
      mp = fmaxf(mp, red[0][w]);
      mb = fmaxf(mb, red[1][w]);
    }
    // non-negative floats order identically as u32 bit patterns
    atomicMax(&cells[0], __float_as_uint(mp));
    atomicMax(&cells[1], __float_as_uint(mb));
  }
}

// ---- kernel 0: zero Ax accumulator + max cells ---------------------------
__global__ void __launch_bounds__(TPB)
init_ws_kernel(float* __restrict__ Ax, unsigned* __restrict__ cells, int m) {
  // fixed per-block slice: no runtime-stride trip-count math
  const int per_block = (m + (int)gridDim.x - 1) / (int)gridDim.x;
  const int lo = blockIdx.x * per_block;
  const int hi = min(lo + per_block, m);
  for (int i = lo + (int)threadIdx.x; i < hi; i += TPB) Ax[i] = 0.0f;
  if (blockIdx.x == 0 && threadIdx.x < 2) cells[threadIdx.x] = 0u;
}

// ---- kernel 3: scalar finalize -------------------------------------------
__global__ void finalize_kernel(const unsigned* __restrict__ cells,
                                float* __restrict__ out) {
  if (threadIdx.x == 0 && blockIdx.x == 0) {
    float part2 = __uint_as_float(cells[0]);
    float part3 = 1.0f + __uint_as_float(cells[1]);
    out[0] = part2 / part3;
  }
}

extern "C" void kernel_launch(void* const* d_in, const int* in_sizes, int n_in,
                              void* d_out, int out_size, void* d_ws, size_t ws_size,
                              hipStream_t stream) {
  const float* A_vals = (const float*)d_in[0];
  const float* b      = (const float*)d_in[1];
  // d_in[2] = c : unused by the reference
  const float* x      = (const float*)d_in[3];
  const float* Iy     = (const float*)d_in[4];
  const int*   A_rows = (const int*)d_in[5];
  const int*   A_cols = (const int*)d_in[6];

  const int nnz = in_sizes[0];
  const int m   = in_sizes[1];

  float*    Ax    = (float*)d_ws;                                        // m floats
  unsigned* cells = (unsigned*)((char*)d_ws + (size_t)m * sizeof(float)); // 2 u32

  init_ws_kernel<<<2048, TPB, 0, stream>>>(Ax, cells, m);

  const int fullTiles = nnz / TILE;
  int g1 = fullTiles > 0 ? (fullTiles < 4096 ? fullTiles : 4096) : 1;
  spmv_scatter_kernel<<<g1, TPB, 0, stream>>>(A_vals, A_rows, A_cols, x, Ax,
                                              nnz, fullTiles);

  proj_max_kernel<<<2048, TPB, 0, stream>>>(Ax, b, Iy, cells, m);
  finalize_kernel<<<1, 64, 0, stream>>>(cells, (float*)d_out);
}